// SpectralRegularization_52948356825705
// MI455X (gfx1250) — compile-verified
//
#include <hip/hip_runtime.h>
#include <math.h>

// ---------------------------------------------------------------------------
// Spectral regularization: out = 1e-4 * (smax(w1) + smax(w2) + smax(w3))
//
// Primary path (needs ~128.5MB workspace):
//   G = W^T W (fp32 WMMA GEMM, 16x16x4), then 8 normalized squarings
//   B_t = (B_{t-1}/s_t)^2 with s_t = maxabs(B_{t-1}) (PSD => s in [l1/n, l1],
//   so l1(B_t) stays in [1, n^2], no fp32 overflow), then
//   l1(B_8) ~= ||B_8||_F^2 / trace(B_8), and
//   ln l1(G) = sum_t 2^-(t-1) ln s_t + 2^-8 ln l1(B_8);  sigma = exp(0.5 ln l1)
//
// Tile movement for the row-major (sN==1) GEMMs uses the CDNA5 Tensor Data
// Mover (tensor_load_to_lds, TENSORcnt) with double-buffered LDS so DMA of
// tile t+1 overlaps WMMA on tile t.
//
// Fallback path (tiny workspace): classic power iteration with W / W^T matvecs.
// ---------------------------------------------------------------------------

typedef __attribute__((ext_vector_type(2))) float v2f;
typedef __attribute__((ext_vector_type(8))) float v8f;
typedef __attribute__((ext_vector_type(4))) unsigned v4u;
typedef __attribute__((ext_vector_type(8))) int v8i;
typedef __attribute__((ext_vector_type(4))) int v4i;

#define TILE 128
#define KT 32
#define LDS_LD (TILE + 4)
#define NSQ 8

// Issue one TDM load of a KT x TILE fp32 tile (row stride sK elements) into
// LDS at byte offset lds_off, with +4-DWORD padding after every 128 DWORDs
// so the LDS image has row pitch LDS_LD floats. Per-wave instruction.
static __device__ __forceinline__ void tdm_load_tile(const float* gptr,
                                                     unsigned lds_off,
                                                     long long sK) {
  unsigned long long ga = (unsigned long long)(size_t)gptr;
  v4u g0;
  g0[0] = 1u;                                        // count=1, load, user mode
  g0[1] = lds_off;                                   // lds_addr (bytes)
  g0[2] = (unsigned)(ga & 0xFFFFFFFFu);              // global_addr[31:0]
  g0[3] = (unsigned)((ga >> 32) & 0x1FFFFFFu)        // global_addr[56:32]
          | (2u << 30);                              // type=2 (image)

  unsigned long long s0 = (unsigned long long)sK;    // dim0 stride (elements)
  v8i g1;
  g1[0] = (int)((2u << 16)                           // data_size = 4B
                | (1u << 20)                         // pad_enable
                | (6u << 22)                         // pad_interval: 128 DWORDs
                | (3u << 25));                       // pad_amount: 4 DWORDs
  g1[1] = (int)(((unsigned)TILE & 0xFFFFu) << 16);   // tensor_dim0[15:0]=128
  g1[2] = (int)(((unsigned)KT & 0xFFFFu) << 16);     // td0[31:16]=0, td1[15:0]=32
  g1[3] = (int)((unsigned)TILE << 16);               // td1[31:16]=0, tile_dim0=128
  g1[4] = (int)((unsigned)KT);                       // tile_dim1=32, tile_dim2=0
  g1[5] = (int)(unsigned)(s0 & 0xFFFFFFFFu);         // dim0_stride[31:0]
  g1[6] = (int)(unsigned)((s0 >> 32) & 0xFFFFu);     // dim0_stride[47:32]
  g1[7] = 0;                                         // dim1_stride (unused, 2D)

  v4i z4 = {};  // D# groups 2/3: zero (2D tensor, features disabled)
  v8i z8 = {};  // trailing int32x8 of the 6-arg toolchain form: zero-filled
  __builtin_amdgcn_tensor_load_to_lds(g0, g1, z4, z4, z8, 0);
}

// ------------------------- WMMA Gram kernel --------------------------------
// C[i][j] = (1/s^2) * sum_k A(k,i) * A(k,j),  i,j in [0,n), k in [0,K)
// A(k,c) = Aptr[k*sK + c*sN].  (sN==1: row-major, TDM path; sK==1: transposed)
__global__ __launch_bounds__(256) void gram_kernel(
    const float* __restrict__ A, float* __restrict__ C,
    int n, int K, long long sK, long long sN, const float* __restrict__ scale) {
  __shared__ float ldsA[2][KT][LDS_LD];
  __shared__ float ldsB[2][KT][LDS_LD];

  const int tid = threadIdx.x;
  const int lane = tid & 31;
  const int wave = tid >> 5;
  const int rgroup = wave >> 1;  // 0..3 -> 32-row band
  const int cgroup = wave & 1;   // 0..1 -> 64-col band
  const int i0 = blockIdx.y * TILE;
  const int j0 = blockIdx.x * TILE;

  v8f acc[2][4] = {};  // 8 x (16x16) fp32 accumulators per wave
  const int kTiles = K / KT;

  auto compute_tile = [&](const float(*tA)[LDS_LD], const float(*tB)[LDS_LD]) {
#pragma unroll
    for (int ks = 0; ks < KT; ks += 4) {
      // f32 16x16x4 fragment layout: lanes 0-15 hold K={ks,ks+1},
      // lanes 16-31 hold K={ks+2,ks+3}; M/N = lane&15 within block.
      const int krow = ks + ((lane >> 4) << 1);
      v2f afrag[2], bfrag[4];
#pragma unroll
      for (int r = 0; r < 2; ++r) {
        int cidx = rgroup * 32 + r * 16 + (lane & 15);
        afrag[r][0] = tA[krow + 0][cidx];
        afrag[r][1] = tA[krow + 1][cidx];
      }
#pragma unroll
      for (int c = 0; c < 4; ++c) {
        int cidx = cgroup * 64 + c * 16 + (lane & 15);
        bfrag[c][0] = tB[krow + 0][cidx];
        bfrag[c][1] = tB[krow + 1][cidx];
      }
#pragma unroll
      for (int r = 0; r < 2; ++r)
#pragma unroll
        for (int c = 0; c < 4; ++c)
          acc[r][c] = __builtin_amdgcn_wmma_f32_16x16x4_f32(
              false, afrag[r], false, bfrag[c], (short)0, acc[r][c], false,
              false);
    }
  };

  if (sN == 1) {
    // -------- TDM double-buffered path --------
    const unsigned offA[2] = {
        (unsigned)(unsigned long long)(size_t)&ldsA[0][0][0],
        (unsigned)(unsigned long long)(size_t)&ldsA[1][0][0]};
    const unsigned offB[2] = {
        (unsigned)(unsigned long long)(size_t)&ldsB[0][0][0],
        (unsigned)(unsigned long long)(size_t)&ldsB[1][0][0]};

    if (wave == 0) {
      tdm_load_tile(A + i0, offA[0], sK);
      tdm_load_tile(A + j0, offB[0], sK);
    }
    for (int kt = 0; kt < kTiles; ++kt) {
      const int buf = kt & 1;
      if (wave == 0) {
        if (kt + 1 < kTiles) {
          const float* p = A + (long long)(kt + 1) * KT * sK;
          tdm_load_tile(p + i0, offA[buf ^ 1], sK);
          tdm_load_tile(p + j0, offB[buf ^ 1], sK);
          __builtin_amdgcn_s_wait_tensorcnt((short)2);  // current tile done
        } else {
          __builtin_amdgcn_s_wait_tensorcnt((short)0);
        }
      }
      __syncthreads();
      compute_tile(ldsA[buf], ldsB[buf]);
      __syncthreads();
    }
  } else {
    // -------- transposed (sK==1) manual path: contiguous along k --------
    for (int kt = 0; kt < kTiles; ++kt) {
      const int kb = kt * KT;
#pragma unroll
      for (int pass = 0; pass < 4; ++pass) {
        int cc = pass * 32 + (tid >> 3);
        int kk = (tid & 7) << 2;
        const float4 va = *(const float4*)(A + (long long)(i0 + cc) * sN + (kb + kk));
        ldsA[0][kk + 0][cc] = va.x; ldsA[0][kk + 1][cc] = va.y;
        ldsA[0][kk + 2][cc] = va.z; ldsA[0][kk + 3][cc] = va.w;
        const float4 vb = *(const float4*)(A + (long long)(j0 + cc) * sN + (kb + kk));
        ldsB[0][kk + 0][cc] = vb.x; ldsB[0][kk + 1][cc] = vb.y;
        ldsB[0][kk + 2][cc] = vb.z; ldsB[0][kk + 3][cc] = vb.w;
      }
      __syncthreads();
      compute_tile(ldsA[0], ldsB[0]);
      __syncthreads();
    }
  }

  float s2inv = 1.0f;
  if (scale) { float s = *scale; s2inv = 1.0f / (s * s); }

#pragma unroll
  for (int r = 0; r < 2; ++r)
#pragma unroll
    for (int c = 0; c < 4; ++c) {
      int baseRow = i0 + rgroup * 32 + r * 16 + ((lane >> 4) << 3);
      int col = j0 + cgroup * 64 + c * 16 + (lane & 15);
#pragma unroll
      for (int e = 0; e < 8; ++e)
        C[(size_t)(baseRow + e) * n + col] = acc[r][c][e] * s2inv;
    }
}

// ------------------------- reductions --------------------------------------
__global__ __launch_bounds__(256) void maxabs_kernel(const float* __restrict__ M,
                                                     size_t count, unsigned* out) {
  __shared__ float red[256];
  size_t stride = (size_t)gridDim.x * 256;
  float m = 0.0f;
  for (size_t i = (size_t)blockIdx.x * 256 + threadIdx.x; i < count; i += stride)
    m = fmaxf(m, fabsf(M[i]));
  red[threadIdx.x] = m;
  __syncthreads();
  for (int s = 128; s > 0; s >>= 1) {
    if (threadIdx.x < (unsigned)s)
      red[threadIdx.x] = fmaxf(red[threadIdx.x], red[threadIdx.x + s]);
    __syncthreads();
  }
  if (threadIdx.x == 0) atomicMax(out, __float_as_uint(red[0]));  // vals >= 0
}

__global__ __launch_bounds__(256) void frob_kernel(const float* __restrict__ M,
                                                   size_t count, float* partials) {
  __shared__ double red[256];
  size_t stride = (size_t)gridDim.x * 256;
  double a = 0.0;
  for (size_t i = (size_t)blockIdx.x * 256 + threadIdx.x; i < count; i += stride) {
    double x = M[i];
    a += x * x;
  }
  red[threadIdx.x] = a;
  __syncthreads();
  for (int s = 128; s > 0; s >>= 1) {
    if (threadIdx.x < (unsigned)s) red[threadIdx.x] += red[threadIdx.x + s];
    __syncthreads();
  }
  if (threadIdx.x == 0) partials[blockIdx.x] = (float)red[0];
}

__global__ __launch_bounds__(256) void sumpart_kernel(const float* __restrict__ p,
                                                      int count, float* out) {
  __shared__ double red[256];
  double a = 0.0;
  for (int i = threadIdx.x; i < count; i += 256) a += (double)p[i];
  red[threadIdx.x] = a;
  __syncthreads();
  for (int s = 128; s > 0; s >>= 1) {
    if (threadIdx.x < (unsigned)s) red[threadIdx.x] += red[threadIdx.x + s];
    __syncthreads();
  }
  if (threadIdx.x == 0) out[0] = (float)red[0];
}

__global__ __launch_bounds__(256) void trace_kernel(const float* __restrict__ B,
                                                    int n, float* out) {
  __shared__ double red[256];
  double a = 0.0;
  for (int i = threadIdx.x; i < n; i += 256) a += (double)B[(size_t)i * n + i];
  red[threadIdx.x] = a;
  __syncthreads();
  for (int s = 128; s > 0; s >>= 1) {
    if (threadIdx.x < (unsigned)s) red[threadIdx.x] += red[threadIdx.x + s];
    __syncthreads();
  }
  if (threadIdx.x == 0) out[0] = (float)red[0];
}

__global__ void init_kernel(float* stats) {
  if (threadIdx.x < 48) stats[threadIdx.x] = 0.0f;
}

// stats layout per matrix (stride 16): [0..7]=s_1..s_8, [8]=trace, [9]=frob^2
__global__ void finalize_kernel(const float* __restrict__ stats, float* out) {
  double total = 0.0;
  for (int m = 0; m < 3; ++m) {
    const float* st = stats + m * 16;
    double loglam = log((double)st[9] / (double)st[8]);  // ln l1(B_p)
    for (int t = NSQ; t >= 1; --t)
      loglam = log((double)st[t - 1]) + 0.5 * loglam;     // ln l1(B_{t-1})
    total += exp(0.5 * loglam);                            // sigma
  }
  out[0] = (float)(1e-4 * total);
}

// ------------------------- fallback (tiny workspace) ------------------------
__global__ __launch_bounds__(256) void matvec_row_kernel(const float* __restrict__ W,
                                                         const float* __restrict__ x,
                                                         float* y, int n) {
  __shared__ float red[256];
  const float* row = W + (size_t)blockIdx.x * n;
  float a = 0.0f;
  for (int c = threadIdx.x; c < n; c += 256) a += row[c] * x[c];
  red[threadIdx.x] = a;
  __syncthreads();
  for (int s = 128; s > 0; s >>= 1) {
    if (threadIdx.x < (unsigned)s) red[threadIdx.x] += red[threadIdx.x + s];
    __syncthreads();
  }
  if (threadIdx.x == 0) y[blockIdx.x] = red[0];
}

__global__ __launch_bounds__(256) void matvec_col_kernel(const float* __restrict__ W,
                                                         const float* __restrict__ x,
                                                         float* y, int m, int n) {
  __shared__ float red[256];
  int c = blockIdx.x;
  float a = 0.0f;
  for (int r = threadIdx.x; r < m; r += 256) a += W[(size_t)r * n + c] * x[r];
  red[threadIdx.x] = a;
  __syncthreads();
  for (int s = 128; s > 0; s >>= 1) {
    if (threadIdx.x < (unsigned)s) red[threadIdx.x] += red[threadIdx.x + s];
    __syncthreads();
  }
  if (threadIdx.x == 0) y[c] = red[0];
}

__global__ void fill_ones_kernel(float* v, int n) {
  int i = blockIdx.x * 256 + threadIdx.x;
  if (i < n) v[i] = 1.0f;
}

__global__ __launch_bounds__(256) void normalize_kernel(float* v, int n, float* norm_out) {
  __shared__ double red[256];
  __shared__ float snorm;
  double a = 0.0;
  for (int i = threadIdx.x; i < n; i += 256) {
    double x = v[i];
    a += x * x;
  }
  red[threadIdx.x] = a;
  __syncthreads();
  for (int s = 128; s > 0; s >>= 1) {
    if (threadIdx.x < (unsigned)s) red[threadIdx.x] += red[threadIdx.x + s];
    __syncthreads();
  }
  if (threadIdx.x == 0) {
    snorm = (float)sqrt(red[0]);
    if (norm_out) *norm_out = snorm;
  }
  __syncthreads();
  float inv = 1.0f / snorm;
  for (int i = threadIdx.x; i < n; i += 256) v[i] *= inv;
}

__global__ void finalize_fb_kernel(const float* __restrict__ stats, float* out) {
  out[0] = 1e-4f * (stats[8] + stats[24] + stats[40]);
}

// ------------------------- host launcher -----------------------------------
extern "C" void kernel_launch(void* const* d_in, const int* in_sizes, int n_in,
                              void* d_out, int out_size, void* d_ws, size_t ws_size,
                              hipStream_t stream) {
  const float* w1 = (const float*)d_in[0];  // 4096 x 4096
  const float* w2 = (const float*)d_in[1];  // 8192 x 2048
  const float* w3 = (const float*)d_in[2];  // 2048 x 8192

  const size_t N2 = (size_t)4096 * 4096;
  const size_t need = (2 * N2 + 1024 + 48) * sizeof(float);

  if (ws_size >= need) {
    float* buf1 = (float*)d_ws;
    float* buf2 = buf1 + N2;
    float* partials = buf2 + N2;
    float* stats = partials + 1024;

    init_kernel<<<1, 64, 0, stream>>>(stats);

    auto run = [&](const float* W, int n, int K, long long sK, long long sN, float* st) {
      dim3 grid(n / TILE, n / TILE);
      gram_kernel<<<grid, 256, 0, stream>>>(W, buf1, n, K, sK, sN, nullptr);
      float* cur = buf1;
      float* oth = buf2;
      for (int t = 0; t < NSQ; ++t) {
        maxabs_kernel<<<1024, 256, 0, stream>>>(cur, (size_t)n * n, (unsigned*)(st + t));
        gram_kernel<<<grid, 256, 0, stream>>>(cur, oth, n, n, (long long)n, 1, st + t);
        float* tmp = cur; cur = oth; oth = tmp;
      }
      trace_kernel<<<1, 256, 0, stream>>>(cur, n, st + 8);
      frob_kernel<<<1024, 256, 0, stream>>>(cur, (size_t)n * n, partials);
      sumpart_kernel<<<1, 256, 0, stream>>>(partials, 1024, st + 9);
    };

    run(w1, 4096, 4096, 4096, 1, stats + 0);
    run(w2, 2048, 8192, 2048, 1, stats + 16);   // G = W2^T W2
    run(w3, 2048, 8192, 1, 8192, stats + 32);   // G = W3 W3^T (transposed access)

    finalize_kernel<<<1, 1, 0, stream>>>(stats, (float*)d_out);
  } else {
    // Fallback: power iteration with W / W^T matvecs (tiny workspace).
    float* u = (float*)d_ws;        // 8192
    float* v = u + 8192;            // 8192
    float* stats = v + 8192;        // 48

    init_kernel<<<1, 64, 0, stream>>>(stats);

    auto runfb = [&](const float* W, int m, int n, float* st) {
      fill_ones_kernel<<<(n + 255) / 256, 256, 0, stream>>>(v, n);
      normalize_kernel<<<1, 256, 0, stream>>>(v, n, nullptr);
      for (int it = 0; it < 400; ++it) {
        matvec_row_kernel<<<m, 256, 0, stream>>>(W, v, u, n);   // u = W v
        matvec_col_kernel<<<n, 256, 0, stream>>>(W, u, v, m, n); // v = W^T u
        normalize_kernel<<<1, 256, 0, stream>>>(v, n, nullptr);
      }
      matvec_row_kernel<<<m, 256, 0, stream>>>(W, v, u, n);
      normalize_kernel<<<1, 256, 0, stream>>>(u, m, st + 8);    // sigma = ||W v||
    };

    runfb(w1, 4096, 4096, stats + 0);
    runfb(w2, 8192, 2048, stats + 16);
    runfb(w3, 2048, 8192, stats + 32);

    finalize_fb_kernel<<<1, 1, 0, stream>>>(stats, (float*)d_out);
  }
}